// GAT_48163763258016
// MI455X (gfx1250) — compile-verified
//
#include <hip/hip_runtime.h>
#include <math.h>

typedef __attribute__((ext_vector_type(2))) float v2f;
typedef __attribute__((ext_vector_type(8))) float v8f;

#define NHEADS 2
#define NEG_SLOPE 0.2f
#define BN_EPS 1e-5f

// ---------------------------------------------------------------- utilities

__global__ void fill_kernel(float* __restrict__ p, long long n, float v) {
    long long i = (long long)blockIdx.x * blockDim.x + threadIdx.x;
    long long stride = (long long)gridDim.x * blockDim.x;
    for (; i < n; i += stride) p[i] = v;
}

__device__ __forceinline__ void atomicMaxF(float* addr, float val) {
    // sign-split trick: correct ordering for mixed-sign floats, init = -inf
    if (val >= 0.0f) atomicMax((int*)addr, __float_as_int(val));
    else             atomicMin((unsigned int*)addr, __float_as_uint(val));
}

__device__ __forceinline__ float lrelu(float x) {
    return x > 0.0f ? x : NEG_SLOPE * x;
}

// ---------------------------------------------------------------- GEMM (WMMA f32)
// H[nrows, NC] = X[nrows, K] @ W[K, NC].  One wave32 per 16x16 output tile.
// A (16xK) and B (Kx16) tiles staged through LDS; inner loop is
// V_WMMA_F32_16X16X4_F32 accumulating fp32.
__global__ void __launch_bounds__(32)
gemm_wmma_f32(const float* __restrict__ X, const float* __restrict__ W,
              float* __restrict__ H, int nrows, int K, int NC) {
    __shared__ float ldsA[16 * 128];   // max K = 128
    __shared__ float ldsB[128 * 16];

    const int m0 = blockIdx.x * 16;
    const int n0 = blockIdx.y * 16;
    if (m0 >= nrows) return;

    const int lane = threadIdx.x;      // 0..31
    const int half = lane >> 4;        // 0 or 1
    const int l    = lane & 15;

    // stage A tile [16 x K], coalesced along K
    for (int r = 0; r < 16; ++r)
        for (int k = lane; k < K; k += 32)
            ldsA[r * K + k] = X[(long long)(m0 + r) * K + k];
    // stage B tile [K x 16]
    for (int k = half; k < K; k += 2)
        ldsB[k * 16 + l] = W[k * NC + n0 + l];
    __syncthreads();

    v8f acc = {0.f, 0.f, 0.f, 0.f, 0.f, 0.f, 0.f, 0.f};
    for (int k0 = 0; k0 < K; k0 += 4) {
        const int ka = k0 + half * 2;  // lanes 0-15: K=k0,k0+1 ; lanes 16-31: K=k0+2,k0+3
        v2f a, b;
        a.x = ldsA[l * K + ka];
        a.y = ldsA[l * K + ka + 1];
        b.x = ldsB[ka * 16 + l];
        b.y = ldsB[(ka + 1) * 16 + l];
        acc = __builtin_amdgcn_wmma_f32_16x16x4_f32(
            /*neg_a=*/false, a, /*neg_b=*/false, b,
            /*c_mod=*/(short)0, acc, /*reuse_a=*/false, /*reuse_b=*/false);
    }

    // C/D layout: VGPR i -> M = i + 8*half ; N = lane&15
    const int Mb = m0 + half * 8;
    const int nc = n0 + l;
#pragma unroll
    for (int i = 0; i < 8; ++i)
        H[(long long)(Mb + i) * NC + nc] = acc[i];
}

// ---------------------------------------------------------------- attention scalars
// asrc/adst[node*2+head] = sum_c H[node, head*C + c] * a{src,dst}[head*C + c]
__global__ void alpha_kernel(const float* __restrict__ H,
                             const float* __restrict__ a_s,
                             const float* __restrict__ a_d,
                             float* __restrict__ asrc, float* __restrict__ adst,
                             int n, int C) {
    int i = blockIdx.x * blockDim.x + threadIdx.x;
    if (i >= n * NHEADS) return;
    int node = i >> 1, head = i & 1;
    const float* hp = H + (long long)node * (NHEADS * C) + head * C;
    float s = 0.f, d = 0.f;
    for (int c = 0; c < C; ++c) {
        float v = hp[c];
        s += v * a_s[head * C + c];
        d += v * a_d[head * C + c];
    }
    asrc[i] = s;
    adst[i] = d;
}

// ---------------------------------------------------------------- edge passes
__device__ __forceinline__ void edge_sd(const int* __restrict__ ei, int e, int E,
                                        int& s, int& d) {
    if (e < E) { s = ei[e]; d = ei[E + e]; }
    else       { s = d = e - E; }           // self loops appended
}

__global__ void edge_max_kernel(const int* __restrict__ ei, int E, int n,
                                const float* __restrict__ asrc,
                                const float* __restrict__ adst,
                                float* __restrict__ mmax) {
    int e = blockIdx.x * blockDim.x + threadIdx.x;
    if (e >= E + n) return;
    int s, d; edge_sd(ei, e, E, s, d);
#pragma unroll
    for (int h = 0; h < NHEADS; ++h) {
        float x = lrelu(asrc[s * NHEADS + h] + adst[d * NHEADS + h]);
        atomicMaxF(&mmax[d * NHEADS + h], x);
    }
}

__global__ void edge_expsum_kernel(const int* __restrict__ ei, int E, int n,
                                   const float* __restrict__ asrc,
                                   const float* __restrict__ adst,
                                   const float* __restrict__ mmax,
                                   float* __restrict__ ssum) {
    int e = blockIdx.x * blockDim.x + threadIdx.x;
    if (e >= E + n) return;
    int s, d; edge_sd(ei, e, E, s, d);
#pragma unroll
    for (int h = 0; h < NHEADS; ++h) {
        float x  = lrelu(asrc[s * NHEADS + h] + adst[d * NHEADS + h]);
        float ex = __expf(x - mmax[d * NHEADS + h]);
        atomicAdd(&ssum[d * NHEADS + h], ex);
    }
}

// one wave per edge; lanes stride the 2*C channels; recompute alpha (L2-hot)
__global__ void edge_scatter_kernel(const int* __restrict__ ei, int E, int n,
                                    const float* __restrict__ H,
                                    const float* __restrict__ asrc,
                                    const float* __restrict__ adst,
                                    const float* __restrict__ mmax,
                                    const float* __restrict__ ssum,
                                    float* __restrict__ agg, int C) {
    int gid  = blockIdx.x * blockDim.x + threadIdx.x;
    int wave = gid >> 5;
    int lane = gid & 31;
    if (wave >= E + n) return;
    int s, d; edge_sd(ei, wave, E, s, d);

    float e0 = lrelu(asrc[s * 2 + 0] + adst[d * 2 + 0]);
    float e1 = lrelu(asrc[s * 2 + 1] + adst[d * 2 + 1]);
    float a0 = __expf(e0 - mmax[d * 2 + 0]) / ssum[d * 2 + 0];
    float a1 = __expf(e1 - mmax[d * 2 + 1]) / ssum[d * 2 + 1];

    const int HC = NHEADS * C;
    const float* hp = H + (long long)s * HC;
    float*       op = agg + (long long)d * HC;
    for (int c = lane; c < HC; c += 32) {
        float alpha = (c < C) ? a0 : a1;
        atomicAdd(&op[c], hp[c] * alpha);
    }
}

// ---------------------------------------------------------------- batch norm (HC == 64)
__global__ void bn_stats_kernel(const float* __restrict__ A,
                                const float* __restrict__ bias,
                                float* __restrict__ stats, int n) {
    __shared__ float ls[2][4][64];
    int c = threadIdx.x & 63, g = threadIdx.x >> 6;
    float sum = 0.f, sq = 0.f;
    for (int r = blockIdx.x * 4 + g; r < n; r += gridDim.x * 4) {
        float v = A[(long long)r * 64 + c] + bias[c];
        sum += v; sq += v * v;
    }
    ls[0][g][c] = sum; ls[1][g][c] = sq;
    __syncthreads();
    if (g == 0) {
        sum = ls[0][0][c] + ls[0][1][c] + ls[0][2][c] + ls[0][3][c];
        sq  = ls[1][0][c] + ls[1][1][c] + ls[1][2][c] + ls[1][3][c];
        atomicAdd(&stats[c], sum);
        atomicAdd(&stats[64 + c], sq);
    }
}

__global__ void bn_apply_kernel(const float* __restrict__ A,
                                const float* __restrict__ bias,
                                const float* __restrict__ gamma,
                                const float* __restrict__ beta,
                                const float* __restrict__ stats,
                                float* __restrict__ out, int n) {
    long long i = (long long)blockIdx.x * blockDim.x + threadIdx.x;
    if (i >= (long long)n * 64) return;
    int c = (int)(i & 63);
    float inv_n = 1.f / (float)n;
    float mu  = stats[c] * inv_n;
    float var = stats[64 + c] * inv_n - mu * mu;
    float v = A[i] + bias[c];
    float y = gamma[c] * (v - mu) * rsqrtf(var + BN_EPS) + beta[c];
    out[i] = y > 0.f ? y : (__expf(y) - 1.f);   // ELU
}

// ---------------------------------------------------------------- layer-2 finalize: head mean + bias
__global__ void finalize_kernel(const float* __restrict__ agg,
                                const float* __restrict__ bias,
                                float* __restrict__ out, int n) {
    long long i = (long long)blockIdx.x * blockDim.x + threadIdx.x;
    if (i >= (long long)n * 40) return;
    int node = (int)(i / 40);
    int c    = (int)(i % 40);
    out[i] = 0.5f * (agg[(long long)node * 80 + c] + agg[(long long)node * 80 + 40 + c]) + bias[c];
}

// ---------------------------------------------------------------- driver

extern "C" void kernel_launch(void* const* d_in, const int* in_sizes, int n_in,
                              void* d_out, int out_size, void* d_ws, size_t ws_size,
                              hipStream_t stream) {
    const float* x   = (const float*)d_in[0];
    const int*   ei  = (const int*)  d_in[1];
    const float* W0  = (const float*)d_in[2];
    const float* as0 = (const float*)d_in[3];
    const float* ad0 = (const float*)d_in[4];
    const float* b0  = (const float*)d_in[5];
    const float* g0  = (const float*)d_in[6];
    const float* bt0 = (const float*)d_in[7];
    const float* W1  = (const float*)d_in[8];
    const float* as1 = (const float*)d_in[9];
    const float* ad1 = (const float*)d_in[10];
    const float* b1  = (const float*)d_in[11];
    const float* g1  = (const float*)d_in[12];
    const float* bt1 = (const float*)d_in[13];
    const float* W2  = (const float*)d_in[14];
    const float* as2 = (const float*)d_in[15];
    const float* ad2 = (const float*)d_in[16];
    const float* b2  = (const float*)d_in[17];

    const int N    = in_sizes[0] / 128;
    const int E    = in_sizes[1] / 2;
    const int Etot = E + N;

    // workspace layout (~67 MB for N=100k, E=1.6M)
    float* A     = (float*)d_ws;               // [N*80] gemm output (H)
    float* B     = A    + (size_t)N * 80;      // [N*80] activations / aggregation
    float* asrc  = B    + (size_t)N * 80;      // [N*2]
    float* adst  = asrc + (size_t)N * 2;       // [N*2]
    float* mmax  = adst + (size_t)N * 2;       // [N*2]
    float* ssum  = mmax + (size_t)N * 2;       // [N*2]
    float* stats = ssum + (size_t)N * 2;       // [128]

    const int egrid = (Etot + 255) / 256;
    const int sgrid = (int)(((long long)Etot * 32 + 255) / 256);

    auto run_layer = [&](const float* in, int K, int C,
                         const float* W, const float* a_s, const float* a_d,
                         const float* bias, const float* gamma, const float* beta,
                         bool has_bn) {
        const int HC = NHEADS * C;
        // GEMM first (it reads B when in == B), then reset accumulators
        dim3 gg((N + 15) / 16, HC / 16);
        gemm_wmma_f32<<<gg, 32, 0, stream>>>(in, W, A, N, K, HC);
        alpha_kernel<<<(N * 2 + 255) / 256, 256, 0, stream>>>(A, a_s, a_d, asrc, adst, N, C);

        fill_kernel<<<1024, 256, 0, stream>>>(B, (long long)N * HC, 0.f);
        fill_kernel<<<256, 256, 0, stream>>>(mmax, (long long)N * 2, -INFINITY);
        fill_kernel<<<256, 256, 0, stream>>>(ssum, (long long)N * 2, 0.f);
        fill_kernel<<<1, 128, 0, stream>>>(stats, 128, 0.f);

        edge_max_kernel   <<<egrid, 256, 0, stream>>>(ei, E, N, asrc, adst, mmax);
        edge_expsum_kernel<<<egrid, 256, 0, stream>>>(ei, E, N, asrc, adst, mmax, ssum);
        edge_scatter_kernel<<<sgrid, 256, 0, stream>>>(ei, E, N, A, asrc, adst, mmax, ssum, B, C);

        if (has_bn) {
            bn_stats_kernel<<<512, 256, 0, stream>>>(B, bias, stats, N);
            bn_apply_kernel<<<(int)(((long long)N * 64 + 255) / 256), 256, 0, stream>>>(
                B, bias, gamma, beta, stats, B, N);
        }
    };

    // layer 0: x[N,128] -> B[N,64] (BN+ELU applied in place)
    run_layer(x, 128, 32, W0, as0, ad0, b0, g0, bt0, true);
    // layer 1: B[N,64] -> B[N,64]
    run_layer(B, 64, 32, W1, as1, ad1, b1, g1, bt1, true);
    // layer 2: B[N,64] -> agg B[N,80] -> mean heads + bias -> out[N,40]
    run_layer(B, 64, 40, W2, as2, ad2, b2, nullptr, nullptr, false);

    finalize_kernel<<<(int)(((long long)N * 40 + 255) / 256), 256, 0, stream>>>(
        B, b2, (float*)d_out, N);
}